// Multihead_Attention_86681029968039
// MI455X (gfx1250) — compile-verified
//
#include <hip/hip_runtime.h>
#include <math.h>

#define Bc   2
#define Sc   384
#define HIDc 512
#define NHc  8
#define Pc   64

typedef __attribute__((ext_vector_type(2))) float v2f;
typedef __attribute__((ext_vector_type(8))) float v8f;

__device__ __forceinline__ v8f wmma_f32(v2f a, v2f b, v8f c) {
  // D = A(16x4 f32) * B(4x16 f32) + C(16x16 f32)
  return __builtin_amdgcn_wmma_f32_16x16x4_f32(false, a, false, b, (short)0, c,
                                               false, false);
}

// ---------------------------------------------------------------------------
// Generic projection: C(MxN) = A(MxK) @ W(KxN) + bias(N), all row-major f32.
// One wave per 16x16 tile.
// ---------------------------------------------------------------------------
__global__ void proj_kernel(const float* __restrict__ A,
                            const float* __restrict__ W,
                            const float* __restrict__ bias,
                            float* __restrict__ C, int M, int N, int K) {
  int n0 = blockIdx.x * 16, m0 = blockIdx.y * 16;
  int lane = threadIdx.x & 31, half = lane >> 4, lm = lane & 15;
  v8f acc = {};
  for (int k0 = 0; k0 < K; k0 += 4) {
    int ka = k0 + half * 2;
    v2f a, b;
    const float* ap = A + (size_t)(m0 + lm) * K + ka;
    a.x = ap[0];
    a.y = ap[1];
    const float* bp = W + (size_t)ka * N + (n0 + lm);
    b.x = bp[0];
    b.y = bp[N];
    acc = wmma_f32(a, b, acc);
  }
  float bv = bias[n0 + lm];
  for (int g = 0; g < 8; ++g)
    C[(size_t)(m0 + g + half * 8) * N + (n0 + lm)] = acc[g] + bv;
}

// ---------------------------------------------------------------------------
// t[b,i,h,c] = sum_p (Q[b,i,hP+p] + vbias[hP+p]) * Wr[c, hP+p]
// Stored as T[((b*S+i)*NH + h)*HID + c].
// ---------------------------------------------------------------------------
__global__ void t_kernel(const float* __restrict__ Q,
                         const float* __restrict__ Wr,
                         const float* __restrict__ vbias,
                         float* __restrict__ T) {
  int c0 = blockIdx.x * 16, m0 = blockIdx.y * 16, h = blockIdx.z;
  int lane = threadIdx.x & 31, half = lane >> 4, lm = lane & 15;
  v8f acc = {};
  for (int p0 = 0; p0 < Pc; p0 += 4) {
    int pa = p0 + half * 2;
    v2f a, b;
    const float* qp = Q + (size_t)(m0 + lm) * HIDc + h * Pc + pa;
    a.x = qp[0] + vbias[h * Pc + pa];
    a.y = qp[1] + vbias[h * Pc + pa + 1];
    // B[p, c] = Wr[c * HID + hP + p]
    const float* wp = Wr + (size_t)(c0 + lm) * HIDc + h * Pc + pa;
    b.x = wp[0];
    b.y = wp[1];
    acc = wmma_f32(a, b, acc);
  }
  for (int g = 0; g < 8; ++g) {
    int m = m0 + g + half * 8;
    T[((size_t)m * NHc + h) * HIDc + (c0 + lm)] = acc[g];
  }
}

// ---------------------------------------------------------------------------
// bconst[m,h] = sum_p br[hP+p] * (Q[m,hP+p] + vbias[hP+p])
// ---------------------------------------------------------------------------
__global__ void bconst_kernel(const float* __restrict__ Q,
                              const float* __restrict__ vbias,
                              const float* __restrict__ br,
                              float* __restrict__ bconst) {
  int idx = blockIdx.x * blockDim.x + threadIdx.x;  // m*NH + h
  if (idx >= Bc * Sc * NHc) return;
  int h = idx & (NHc - 1), m = idx >> 3;
  float s = 0.f;
  for (int p = 0; p < Pc; ++p)
    s += br[h * Pc + p] * (Q[(size_t)m * HIDc + h * Pc + p] + vbias[h * Pc + p]);
  bconst[idx] = s;
}

// ---------------------------------------------------------------------------
// scores[b,h,i,j] = sum_p (Q[b,i,hP+p] + u[hP+p]) * K[b,j,hP+p]   (A_C term)
// ---------------------------------------------------------------------------
__global__ void ac_kernel(const float* __restrict__ Q,
                          const float* __restrict__ Km,
                          const float* __restrict__ ubias,
                          float* __restrict__ scores) {
  int j0 = blockIdx.x * 16, i0 = blockIdx.y * 16;
  int bh = blockIdx.z;
  int b = bh >> 3, h = bh & 7;
  int lane = threadIdx.x & 31, half = lane >> 4, lm = lane & 15;
  const float* Qb = Q + (size_t)b * Sc * HIDc;
  const float* Kb = Km + (size_t)b * Sc * HIDc;
  v8f acc = {};
  for (int p0 = 0; p0 < Pc; p0 += 4) {
    int pa = p0 + half * 2;
    v2f a, bf;
    const float* qp = Qb + (size_t)(i0 + lm) * HIDc + h * Pc + pa;
    a.x = qp[0] + ubias[h * Pc + pa];
    a.y = qp[1] + ubias[h * Pc + pa + 1];
    // B[p, j] = K[b, j, hP+p]
    const float* kp = Kb + (size_t)(j0 + lm) * HIDc + h * Pc + pa;
    bf.x = kp[0];
    bf.y = kp[1];
    acc = wmma_f32(a, bf, acc);
  }
  float* S = scores + ((size_t)bh * Sc + i0) * Sc;
  for (int g = 0; g < 8; ++g)
    S[(size_t)(g + half * 8) * Sc + (j0 + lm)] = acc[g];
}

// ---------------------------------------------------------------------------
// Streaming B_D term: one workgroup per (b,i); 8 waves cover 24 j-tiles.
// D[j,h] = sum_c rel_pos[b,i,j,c] * t[b,i,h,c];  scores += D + bconst.
// rel_pos (604 MB total) is streamed through LDS with double-buffered
// global_load_async_to_lds_b128 (ASYNCcnt pipeline); t slab lives in LDS.
// ---------------------------------------------------------------------------
#define CHUNKF 32                 // floats of c per chunk (128 B per row)
#define NCHUNK (HIDc / CHUNKF)    // 16 chunks per j-tile
#define RSTRIDE 36                // padded row stride in floats (bank-conflict-free)
#define BUF_FLOATS (16 * RSTRIDE) // 576 floats per chunk buffer
#define TSTRIDE 516               // padded t-row stride in floats

__global__ void bd_kernel(const float* __restrict__ rel,
                          const float* __restrict__ T,
                          const float* __restrict__ bconst,
                          float* __restrict__ scores) {
  __shared__ float tbuf[NHc * TSTRIDE];            // 16512 B
  __shared__ float abuf[8 * 2 * BUF_FLOATS];       // 36864 B

  int bi = blockIdx.x;  // b*S + i
  int b = bi / Sc, i = bi % Sc;
  int tid = threadIdx.x;
  int wave = tid >> 5;
  int lane = tid & 31, half = lane >> 4, lm = lane & 15;
  const float* Arow = rel + (size_t)bi * Sc * HIDc;  // [j][c]
  const float* Tg = T + (size_t)bi * NHc * HIDc;     // [h][c]

  // cooperative, coalesced stage of t[b,i] into LDS (padded rows)
  for (int idx = tid; idx < NHc * HIDc; idx += 256) {
    int h = idx >> 9, c = idx & (HIDc - 1);
    tbuf[h * TSTRIDE + c] = Tg[idx];
  }
  __syncthreads();

  float hmask = (lm < NHc) ? 1.f : 0.f;
  int hidx = (lm < NHc) ? lm : 0;

  // async-copy lane mapping: 4 rows x 8 segments of 16B per issue;
  // 4 issues cover a 16-row x 128B chunk, one full cache line per row.
  int rr = lane >> 3;       // 0..3
  int seg = lane & 7;       // 0..7
  float* mybuf = abuf + wave * 2 * BUF_FLOATS;
  unsigned lds0 = (unsigned)(size_t)mybuf + rr * (RSTRIDE * 4) + seg * 16;

  for (int jt = wave; jt < Sc / 16; jt += 8) {
    int j0 = jt * 16;
    unsigned long long gbase =
        (unsigned long long)(Arow + (size_t)(j0 + rr) * HIDc) + seg * 16;

    // prologue: chunk 0 -> buffer 0
#pragma unroll
    for (int it = 0; it < 4; ++it) {
      unsigned l = lds0 + it * (4 * RSTRIDE * 4);
      unsigned long long g = gbase + (unsigned long long)it * 4 * HIDc * 4;
      asm volatile("global_load_async_to_lds_b128 %0, %1, off"
                   :: "v"(l), "v"(g) : "memory");
    }

    v8f acc = {};
    for (int k = 0; k < NCHUNK; ++k) {
      int cur = k & 1;
      if (k < NCHUNK - 1) {  // issue chunk k+1 into the other buffer
        unsigned lnext = lds0 + (1 - cur) * (BUF_FLOATS * 4);
        unsigned long long gnext =
            gbase + (unsigned long long)(k + 1) * (CHUNKF * 4);
#pragma unroll
        for (int it = 0; it < 4; ++it) {
          unsigned l = lnext + it * (4 * RSTRIDE * 4);
          unsigned long long g = gnext + (unsigned long long)it * 4 * HIDc * 4;
          asm volatile("global_load_async_to_lds_b128 %0, %1, off"
                       :: "v"(l), "v"(g) : "memory");
        }
        asm volatile("s_wait_asynccnt 4" ::: "memory");  // chunk k landed
      } else {
        asm volatile("s_wait_asynccnt 0" ::: "memory");  // last chunk landed
      }

      const float* buf = mybuf + cur * BUF_FLOATS;
#pragma unroll
      for (int cc = 0; cc < CHUNKF; cc += 4) {
        int ca = cc + half * 2;
        v2f a, bf;
        a.x = buf[lm * RSTRIDE + ca];
        a.y = buf[lm * RSTRIDE + ca + 1];
        int cg = k * CHUNKF + ca;  // absolute c for the t slab
        bf.x = tbuf[hidx * TSTRIDE + cg] * hmask;
        bf.y = tbuf[hidx * TSTRIDE + cg + 1] * hmask;
        acc = wmma_f32(a, bf, acc);
      }
    }

    if (lm < NHc) {
      int h = lm;
      float bc = bconst[(size_t)bi * NHc + h];
      float* S = scores + (((size_t)(b * NHc + h)) * Sc + i) * Sc;
      for (int g = 0; g < 8; ++g) {
        int j = j0 + g + half * 8;
        S[j] += acc[g] + bc;
      }
    }
  }
}

// ---------------------------------------------------------------------------
// Row softmax with mask + 1/sqrt(P) scale. One wave32 per (b,h,i) row.
// ---------------------------------------------------------------------------
__global__ void softmax_kernel(float* __restrict__ scores,
                               const unsigned char* __restrict__ mask) {
  int row = blockIdx.x;  // (b*NH+h)*S + i
  int i = row % Sc;
  int bh = row / Sc;
  int b = bh >> 3;
  float* S = scores + (size_t)row * Sc;
  const unsigned char* Mrow = mask + ((size_t)b * Sc + i) * Sc;
  int lane = threadIdx.x;
  float vals[Sc / 32];
  float mx = -INFINITY;
  for (int t = 0; t < Sc / 32; ++t) {
    int j = lane + t * 32;
    float s = S[j] * 0.125f;  // 1/sqrt(64)
    s = Mrow[j] ? s : -1e15f;
    vals[t] = s;
    mx = fmaxf(mx, s);
  }
  for (int off = 16; off > 0; off >>= 1) mx = fmaxf(mx, __shfl_xor(mx, off, 32));
  float sum = 0.f;
  for (int t = 0; t < Sc / 32; ++t) {
    vals[t] = __expf(vals[t] - mx);
    sum += vals[t];
  }
  for (int off = 16; off > 0; off >>= 1) sum += __shfl_xor(sum, off, 32);
  float inv = 1.f / sum;
  for (int t = 0; t < Sc / 32; ++t) S[lane + t * 32] = vals[t] * inv;
}

// ---------------------------------------------------------------------------
// out[b,i,hP+p] = sum_j attn[b,h,i,j] * V[b,j,hP+p]
// ---------------------------------------------------------------------------
__global__ void av_kernel(const float* __restrict__ attn,
                          const float* __restrict__ V,
                          float* __restrict__ out) {
  int p0 = blockIdx.x * 16, i0 = blockIdx.y * 16;
  int bh = blockIdx.z;
  int b = bh >> 3, h = bh & 7;
  int lane = threadIdx.x & 31, half = lane >> 4, lm = lane & 15;
  const float* A = attn + (size_t)bh * Sc * Sc;
  const float* Vb = V + (size_t)b * Sc * HIDc;
  v8f acc = {};
  for (int j0 = 0; j0 < Sc; j0 += 4) {
    int ja = j0 + half * 2;
    v2f a, bf;
    const float* ap = A + (size_t)(i0 + lm) * Sc + ja;
    a.x = ap[0];
    a.y = ap[1];
    const float* vp = Vb + (size_t)ja * HIDc + h * Pc + p0 + lm;
    bf.x = vp[0];
    bf.y = vp[HIDc];
    acc = wmma_f32(a, bf, acc);
  }
  float* O = out + ((size_t)(b * Sc) + i0) * HIDc + h * Pc + p0;
  for (int g = 0; g < 8; ++g)
    O[(size_t)(g + half * 8) * HIDc + lm] = acc[g];
}

// ---------------------------------------------------------------------------
extern "C" void kernel_launch(void* const* d_in, const int* in_sizes, int n_in,
                              void* d_out, int out_size, void* d_ws,
                              size_t ws_size, hipStream_t stream) {
  (void)in_sizes; (void)n_in; (void)out_size; (void)ws_size;
  const float* feature = (const float*)d_in[0];
  const float* rel_pos = (const float*)d_in[1];
  const unsigned char* mask = (const unsigned char*)d_in[2];
  const float* Wq = (const float*)d_in[3];
  const float* bq = (const float*)d_in[4];
  const float* Wk = (const float*)d_in[5];
  const float* bk = (const float*)d_in[6];
  const float* Wv = (const float*)d_in[7];
  const float* bv = (const float*)d_in[8];
  const float* Wr = (const float*)d_in[9];
  const float* br = (const float*)d_in[10];
  const float* u  = (const float*)d_in[11];
  const float* v  = (const float*)d_in[12];
  const float* Wf = (const float*)d_in[13];
  const float* bf = (const float*)d_in[14];
  float* out = (float*)d_out;

  const int M = Bc * Sc;  // 768
  float* ws = (float*)d_ws;
  float* Qw = ws;                          // 393216
  float* Kw = Qw + (size_t)M * HIDc;       // 393216
  float* Vw = Kw + (size_t)M * HIDc;       // 393216
  float* Tw = Vw + (size_t)M * HIDc;       // 3145728
  float* bc = Tw + (size_t)M * NHc * HIDc; // 6144
  float* sc = bc + (size_t)M * NHc;        // 2359296
  float* ao = sc + (size_t)Bc * NHc * Sc * Sc;  // 393216

  dim3 w32(32);
  // Q / K / V projections
  proj_kernel<<<dim3(HIDc / 16, M / 16), w32, 0, stream>>>(feature, Wq, bq, Qw, M, HIDc, HIDc);
  proj_kernel<<<dim3(HIDc / 16, M / 16), w32, 0, stream>>>(feature, Wk, bk, Kw, M, HIDc, HIDc);
  proj_kernel<<<dim3(HIDc / 16, M / 16), w32, 0, stream>>>(feature, Wv, bv, Vw, M, HIDc, HIDc);
  // t tensor + bias constant (folds Wr/br through q+v, avoiding r materialization)
  t_kernel<<<dim3(HIDc / 16, M / 16, NHc), w32, 0, stream>>>(Qw, Wr, v, Tw);
  bconst_kernel<<<dim3((M * NHc + 255) / 256), dim3(256), 0, stream>>>(Qw, v, br, bc);
  // A_C term into scores
  ac_kernel<<<dim3(Sc / 16, Sc / 16, Bc * NHc), w32, 0, stream>>>(Qw, Kw, u, sc);
  // streaming B_D term (reads rel_pos exactly once, async-LDS pipelined)
  bd_kernel<<<dim3(Bc * Sc), dim3(256), 0, stream>>>(rel_pos, Tw, bc, sc);
  // masked scaled softmax
  softmax_kernel<<<dim3(Bc * NHc * Sc), w32, 0, stream>>>(sc, mask);
  // attn @ V
  av_kernel<<<dim3(Pc / 16, Sc / 16, Bc * NHc), w32, 0, stream>>>(sc, Vw, ao);
  // final projection
  proj_kernel<<<dim3(HIDc / 16, M / 16), w32, 0, stream>>>(ao, Wf, bf, out, M, HIDc, HIDc);
}